// AdaptiveTokenSampling_26680336843007
// MI455X (gfx1250) — compile-verified
//
#include <hip/hip_runtime.h>
#include <hip/hip_bf16.h>
#include <float.h>

typedef __attribute__((ext_vector_type(2))) float v2f;
typedef __attribute__((ext_vector_type(8))) float v8f;

#define Bc   8
#define Hc   12
#define Nc   1025
#define Dc   64
#define Kc   256
#define NM1  1024
#define KP1  257
#define EPSF 1e-6f

// ---------------------------------------------------------------------------
// Kernel 1: squared norms via WMMA Gram-diagonal.
// One wave (32 lanes) handles 16 tokens x 64 features.
// A (16x4 f32) layout: lane L holds M = L%16; VGPR0/1 = K pair; lanes 16-31
// hold the K+2/K+3 pair. For B = A^T (Gram), the per-lane B data equals the
// per-lane A data, so we feed the same registers to both operands.
// D (16x16 f32): element (M,N) at lane N + 16*(M>=8), vgpr M%8.
// Diagonal t: t<8 -> lane t, c[t];  t>=8 -> lane t+16, c[t-8].
// ---------------------------------------------------------------------------
__global__ __launch_bounds__(32)
void norms_kernel(const float* __restrict__ value, float* __restrict__ norms) {
    const int tile = blockIdx.x;        // 0..63  (16 tokens each)
    const int bh   = blockIdx.y;        // 0..B*H-1
    const int lane = threadIdx.x;
    const int tok  = lane & 15;
    const int half = lane >> 4;

    const int token = 1 + tile * 16 + tok;              // skip CLS token
    const float* row = value + ((size_t)bh * Nc + token) * Dc;

    v2f a[16];
#pragma unroll
    for (int s = 0; s < 16; ++s) {
        const int k0 = 4 * s + 2 * half;
        a[s][0] = row[k0];
        a[s][1] = row[k0 + 1];
    }

    v8f c = {0.f, 0.f, 0.f, 0.f, 0.f, 0.f, 0.f, 0.f};
#pragma unroll
    for (int s = 0; s < 16; ++s) {
        c = __builtin_amdgcn_wmma_f32_16x16x4_f32(
                /*neg_a=*/false, a[s], /*neg_b=*/false, a[s],
                /*c_mod=*/(short)0, c, /*reuse_a=*/false, /*reuse_b=*/false);
    }

    // extract diagonal (each lane owns at most one token's squared norm)
    int t = -1; float sq = 0.f;
    if (half == 0 && tok < 8)  { t = tok; sq = c[tok]; }
    if (half == 1 && tok >= 8) { t = tok; sq = c[tok - 8]; }
    if (t >= 0)
        norms[(size_t)bh * NM1 + tile * 16 + t] = sqrtf(sq);
}

// ---------------------------------------------------------------------------
// Kernel 2: cls_score, normalization, log, mask -> pseudo_logits per batch.
// ---------------------------------------------------------------------------
__global__ __launch_bounds__(256)
void score_kernel(const float* __restrict__ attn,
                  const unsigned char* __restrict__ mask,
                  const float* __restrict__ norms,
                  float* __restrict__ logits) {
    const int b   = blockIdx.x;
    const int tid = threadIdx.x;
    __shared__ float red[256];

    float sc[4];
    float lsum = 0.f;
#pragma unroll
    for (int q = 0; q < 4; ++q) {
        const int j = tid + q * 256;                    // 0..1023
        float s = 0.f;
#pragma unroll
        for (int h = 0; h < Hc; ++h) {
            const size_t bh = (size_t)b * Hc + h;
            const float a  = attn[(bh * Nc + 0) * Nc + 1 + j];  // attn[b,h,0,1+j]
            const float nv = norms[bh * NM1 + j];
            s += a * nv;
        }
        sc[q] = s;
        lsum += s;
    }

    red[tid] = lsum;
    __syncthreads();
    for (int off = 128; off > 0; off >>= 1) {
        if (tid < off) red[tid] += red[tid + off];
        __syncthreads();
    }
    const float denom = red[0] + EPSF;

    const float mask_value = -FLT_MAX * 0.5f;
#pragma unroll
    for (int q = 0; q < 4; ++q) {
        const int j = tid + q * 256;
        const float pl = logf(sc[q] / denom + EPSF);
        const bool m = mask[(size_t)b * Nc + 1 + j] != 0;
        logits[(size_t)b * NM1 + j] = m ? pl : mask_value;
    }
}

// ---------------------------------------------------------------------------
// Kernel 3: argmax over 1024 logits+gumbel per (b,k); one wave each.
// First-occurrence tie-break to match jnp.argmax.
// ---------------------------------------------------------------------------
__global__ __launch_bounds__(32)
void argmax_kernel(const float* __restrict__ gumbel,
                   const float* __restrict__ logits,
                   int* __restrict__ sampled) {
    const int kk   = blockIdx.x;
    const int b    = blockIdx.y;
    const int lane = threadIdx.x;

    const float* g  = gumbel + ((size_t)b * Kc + kk) * NM1;
    const float* pl = logits + (size_t)b * NM1;

    float best = -FLT_MAX;
    int   bidx = 0x7FFFFFFF;
    for (int j = lane; j < NM1; j += 32) {
        const float v = pl[j] + g[j];
        if (v > best) { best = v; bidx = j; }
    }
#pragma unroll
    for (int off = 16; off > 0; off >>= 1) {
        const float ov = __shfl_down(best, off, 32);
        const int   oi = __shfl_down(bidx, off, 32);
        if (ov > best || (ov == best && oi < bidx)) { best = ov; bidx = oi; }
    }
    if (lane == 0) sampled[b * Kc + kk] = bidx + 1;
}

// ---------------------------------------------------------------------------
// Kernel 4: per-batch bitonic sort + dedupe (dup -> BIG -> resort -> 0),
// emit unique_ids (int, ws) and new_mask / unique_ids (float, d_out tail).
// ---------------------------------------------------------------------------
__device__ __forceinline__ void bitonic_sort_256(int* sh, int tid) {
    for (int ksz = 2; ksz <= Kc; ksz <<= 1) {
        for (int j = ksz >> 1; j > 0; j >>= 1) {
            const int ixj = tid ^ j;
            if (ixj > tid) {
                const int x = sh[tid];
                const int y = sh[ixj];
                const bool up = ((tid & ksz) == 0);
                if ((x > y) == up) { sh[tid] = y; sh[ixj] = x; }
            }
            __syncthreads();
        }
    }
}

__global__ __launch_bounds__(256)
void sort_kernel(const int* __restrict__ sampled,
                 int* __restrict__ unique,
                 float* __restrict__ out_mask,
                 float* __restrict__ out_ids) {
    const int b   = blockIdx.x;
    const int tid = threadIdx.x;
    __shared__ int sh[Kc];

    sh[tid] = sampled[b * Kc + tid];
    __syncthreads();

    bitonic_sort_256(sh, tid);

    int v = sh[tid];
    const int prev = (tid > 0) ? sh[tid - 1] : -1;
    __syncthreads();
    if (tid > 0 && v == prev) v = Nc;   // BIG = n
    sh[tid] = v;
    __syncthreads();

    bitonic_sort_256(sh, tid);

    int s = sh[tid];
    if (s == Nc) s = 0;

    unique[b * KP1 + 1 + tid]   = s;
    out_mask[b * KP1 + 1 + tid] = (s != 0) ? 1.0f : 0.0f;
    out_ids[b * KP1 + 1 + tid]  = (float)s;
    if (tid == 0) {
        unique[b * KP1]   = 0;
        out_mask[b * KP1] = 1.0f;   // padded True
        out_ids[b * KP1]  = 0.0f;   // padded 0
    }
}

// ---------------------------------------------------------------------------
// Kernel 5: row gather new_attn[b,h,r,:] = attn[b,h,unique[b,r],:].
// One block per output row; coalesced b32 copy; prefetch the source row.
// ---------------------------------------------------------------------------
__global__ __launch_bounds__(256)
void gather_kernel(const float* __restrict__ attn,
                   const int* __restrict__ unique,
                   float* __restrict__ out) {
    const int r  = blockIdx.x;          // 0..256
    const int bh = blockIdx.y;          // 0..B*H-1
    const int b  = bh / Hc;
    const int tid = threadIdx.x;

    const int id = unique[b * KP1 + r];
    const float* src = attn + ((size_t)bh * Nc + id) * Nc;
    float* dst = out + ((size_t)bh * KP1 + r) * Nc;

    // gfx1250: global_prefetch_b8 the source row (4 floats per lane covers it)
    __builtin_prefetch(src + tid * 4, 0, 0);

    for (int j = tid; j < Nc; j += 256)
        dst[j] = src[j];
}

// ---------------------------------------------------------------------------
extern "C" void kernel_launch(void* const* d_in, const int* in_sizes, int n_in,
                              void* d_out, int out_size, void* d_ws, size_t ws_size,
                              hipStream_t stream) {
    const float*         attn   = (const float*)d_in[0];          // (B,H,N,N)
    const float*         value  = (const float*)d_in[1];          // (B,H,N,D)
    const unsigned char* mask   = (const unsigned char*)d_in[2];  // (B,N) bool
    const float*         gumbel = (const float*)d_in[3];          // (B,K,N-1)

    float* out      = (float*)d_out;
    float* out_attn = out;                                        // B*H*KP1*N
    float* out_mask = out + (size_t)Bc * Hc * KP1 * Nc;           // B*KP1
    float* out_ids  = out_mask + (size_t)Bc * KP1;                // B*KP1

    float* norms  = (float*)d_ws;                                 // B*H*NM1
    float* logits = norms + (size_t)Bc * Hc * NM1;                // B*NM1
    int*   sampled = (int*)(logits + (size_t)Bc * NM1);           // B*K
    int*   unique  = sampled + (size_t)Bc * Kc;                   // B*KP1

    norms_kernel <<<dim3(NM1 / 16, Bc * Hc), 32,  0, stream>>>(value, norms);
    score_kernel <<<dim3(Bc),               256, 0, stream>>>(attn, mask, norms, logits);
    argmax_kernel<<<dim3(Kc, Bc),            32, 0, stream>>>(gumbel, logits, sampled);
    sort_kernel  <<<dim3(Bc),               256, 0, stream>>>(sampled, unique, out_mask, out_ids);
    gather_kernel<<<dim3(KP1, Bc * Hc),     256, 0, stream>>>(attn, unique, out_attn);
}